// BidirectionalLayerNeural_47373489274951
// MI455X (gfx1250) — compile-verified
//
#include <hip/hip_runtime.h>

#define NPTS 8192
#define C0   64
#define E    128
#define KNN  16
#define NT   16   // rows per workgroup in fused kernel
#define MT   64   // candidate columns per tile
#define ROWW (E + 4)    // padded LDS row stride (floats) for g tiles
#define DSW  (MT + 4)   // padded LDS row stride for distance tile

typedef __attribute__((ext_vector_type(2))) float v2f;
typedef __attribute__((ext_vector_type(8))) float v8f;
typedef __attribute__((ext_vector_type(4))) unsigned int v4u;
typedef __attribute__((ext_vector_type(8))) int v8i_;
typedef __attribute__((ext_vector_type(4))) int v4i_;

// --- fused-kernel LDS layout (bytes), single dynamic segment -----------------
#define L_G1S   0                                   // NT*ROWW*4      = 8448
#define L_G2S0  (L_G1S  + NT * ROWW * 4)            // 64*132*4       = 33792
#define L_G2S1  (L_G2S0 + MT * ROWW * 4)
#define L_P1S   (L_G2S1 + MT * ROWW * 4)            // 16 * float4    = 256
#define L_P2S0  (L_P1S  + NT * 16)                  // 64 * float4    = 1024
#define L_P2S1  (L_P2S0 + MT * 16)
#define L_DS    (L_P2S1 + MT * 16)                  // NT*DSW*4       = 4352
#define L_MRGD  (L_DS   + NT * DSW * 4)             // NT*4*16*4      = 4096
#define L_MRGI  (L_MRGD + NT * 4 * KNN * 4)
#define L_TOTAL (L_MRGI + NT * 4 * KNN * 4)         // = 90880 bytes

// numeric LDS byte offset of a generic pointer into shared memory
__device__ __forceinline__ unsigned lds_addr_of(const void* p) {
    return (unsigned)(unsigned long long)
        (__attribute__((address_space(3))) const char*)p;
}

// Issue one TDM 2D tile load: MT rows x E floats from g (row stride E floats)
// into LDS at lds_addr, with HW padding of 4 DWORDs after every 128 DWORDs
// (reproduces the ROWW=E+4 bank-conflict padding).
__device__ __forceinline__ void tdm_load_g2_tile(const float* gsrc,
                                                 unsigned lds_addr) {
    const unsigned long long ga = (unsigned long long)gsrc;
    v4u g0;
    g0[0] = 1u;                                   // count=1, user mode
    g0[1] = lds_addr;                             // lds_addr [63:32]
    g0[2] = (unsigned)(ga & 0xffffffffu);         // global_addr lo
    g0[3] = (unsigned)((ga >> 32) & 0x01ffffffu)  // global_addr hi (57-bit)
          | (2u << 30);                           // type = 2 (image)
    v8i_ g1;
    g1[0] = (int)((2u << 16)      // data_size = 4 bytes
          |      (1u << 20)       // pad_enable
          |      (6u << 22)       // pad_interval: 128 DWORDs
          |      (3u << 25));     // pad_amount:   4 DWORDs
    g1[1] = (int)(128u << 16);    // tensor_dim0 = 128 (lo16 @ bits 63:48)
    g1[2] = (int)(8192u << 16);   // tensor_dim1 = 8192 (lo16 @ bits 95:80)
    g1[3] = (int)(128u << 16);    // tile_dim0 = 128 (bits 127:112)
    g1[4] = (int)MT;              // tile_dim1 = 64 rows
    g1[5] = (int)E;               // tensor_dim0_stride = 128
    g1[6] = 0;
    g1[7] = 0;
    const v4i_ z4 = {0, 0, 0, 0};           // groups 2/3 unused (2D tensor)
    const v8i_ z8 = {0, 0, 0, 0, 0, 0, 0, 0};
    __builtin_amdgcn_tensor_load_to_lds(g0, g1, z4, z4, z8, 0);
}

// ---------------------------------------------------------------------------
// Phase 0: feature projections  q[which][n][o] = sum_c w[o,c]*feat[c,n] + b[o]
// which: 0=t11(feat1), 1=t22(feat2), 2=t11(feat2), 3=t22(feat1)
// ---------------------------------------------------------------------------
__global__ __launch_bounds__(256) void proj_kernel(
    const float* __restrict__ feat1, const float* __restrict__ feat2,
    const float* __restrict__ t11_w, const float* __restrict__ t11_b,
    const float* __restrict__ t22_w, const float* __restrict__ t22_b,
    float* __restrict__ q)
{
    const int which = blockIdx.z;
    const float* feat = (which == 0 || which == 3) ? feat1 : feat2;
    const float* w    = (which == 0 || which == 2) ? t11_w : t22_w;
    const float* b    = (which == 0 || which == 2) ? t11_b : t22_b;

    const int tid = threadIdx.x;
    const int n = blockIdx.x * 4 + (tid >> 6);
    const int o = tid & 63;

    float acc = b[o];
    #pragma unroll 8
    for (int c = 0; c < C0; ++c)
        acc += w[o * C0 + c] * feat[c * NPTS + n];
    q[((size_t)which * NPTS + n) * C0 + o] = acc;
}

// ---------------------------------------------------------------------------
// Phase 1: normalized embedding  g[which][n][e] = normalize_e(q[n]·dist_w[e]+b)
// ---------------------------------------------------------------------------
__global__ __launch_bounds__(128) void gvec_kernel(
    const float* __restrict__ q,
    const float* __restrict__ dist_w, const float* __restrict__ dist_b,
    float* __restrict__ g)
{
    const int which = blockIdx.y;
    const int n = blockIdx.x;
    const int e = threadIdx.x;

    __shared__ float qs[C0];
    __shared__ float red[128];

    const float* qn = q + ((size_t)which * NPTS + n) * C0;
    if (e < C0) qs[e] = qn[e];
    __syncthreads();

    float acc = dist_b[e];
    #pragma unroll 8
    for (int c = 0; c < C0; ++c)
        acc += dist_w[e * C0 + c] * qs[c];

    red[e] = acc * acc;
    __syncthreads();
    for (int s = 64; s > 0; s >>= 1) {
        if (e < s) red[e] += red[e + s];
        __syncthreads();
    }
    const float inv = 1.0f / (sqrtf(red[0]) + 1e-8f);
    g[((size_t)which * NPTS + n) * E + e] = acc * inv;
}

// ---------------------------------------------------------------------------
// Phase 2 (dominant): fused neural-distance GEMM (fp32 WMMA) + running top-16,
// with TDM (tensor_load_to_lds) double-buffered candidate tiles.
// 128 threads = 4 waves; 16 rows/WG; 64-column tiles; E=128 via 32 chained
// v_wmma_f32_16x16x4_f32 per wave per tile.
// ---------------------------------------------------------------------------
__global__ __launch_bounds__(128) void dist_topk_kernel(
    const float* __restrict__ g1,   // [NPTS][E]
    const float* __restrict__ g2,   // [NPTS][E]
    const float* __restrict__ x1,   // [3][NPTS]
    const float* __restrict__ x2,   // [3][NPTS]
    int* __restrict__ idx_out)      // [NPTS][KNN]
{
    extern __shared__ char smem[];
    float*  g1s  = (float*)(smem + L_G1S);    // [NT][ROWW]
    float*  g2sA = (float*)(smem + L_G2S0);   // [MT][ROWW] double buffer
    float*  g2sB = (float*)(smem + L_G2S1);
    float4* p1s  = (float4*)(smem + L_P1S);   // [NT]
    float4* p2sA = (float4*)(smem + L_P2S0);  // [MT] double buffer
    float4* p2sB = (float4*)(smem + L_P2S1);
    float*  ds   = (float*)(smem + L_DS);     // [NT][DSW]
    float*  mrgd = (float*)(smem + L_MRGD);   // [NT][4][KNN]
    int*    mrgi = (int*)  (smem + L_MRGI);

    const int tid  = threadIdx.x;
    const int wave = tid >> 5;
    const int lane = tid & 31;
    const int n0   = blockIdx.x * NT;

    const unsigned ldsA = lds_addr_of(g2sA);
    const unsigned ldsB = lds_addr_of(g2sB);

    // prologue: stage g1 rows + p1, kick TDM for candidate tile 0, stage p2[0]
    if (wave == 0) tdm_load_g2_tile(g2, ldsA);
    for (int i = tid; i < NT * E; i += 128) {
        const int r = i >> 7, c = i & 127;
        g1s[r * ROWW + c] = g1[(size_t)(n0 + r) * E + c];
    }
    if (tid < NT) {
        const float a = x1[0 * NPTS + n0 + tid];
        const float b = x1[1 * NPTS + n0 + tid];
        const float c = x1[2 * NPTS + n0 + tid];
        p1s[tid] = make_float4(a, b, c, a * a + b * b + c * c);
    }
    if (tid < MT) {
        const float a = x2[0 * NPTS + tid];
        const float b = x2[1 * NPTS + tid];
        const float c = x2[2 * NPTS + tid];
        p2sA[tid] = make_float4(a, b, c, a * a + b * b + c * c);
    }

    // per-owner sorted top-K in registers (owners: tid<64, 4 per row)
    float bd[KNN];
    int   bi[KNN];
    #pragma unroll
    for (int i = 0; i < KNN; ++i) { bd[i] = 3.0e38f; bi[i] = -1; }

    const int row_a = lane & 15;          // A row / C column within tile
    const int khalf = lane >> 4;
    const int mloc  = wave * 16 + row_a;  // local candidate column

    const int NTILES = NPTS / MT;
    for (int t = 0; t < NTILES; ++t) {
        const int m0  = t * MT;
        const int cur = t & 1;
        const float*  gb = cur ? g2sB : g2sA;
        const float4* pb = cur ? p2sB : p2sA;

        if (wave == 0) __builtin_amdgcn_s_wait_tensorcnt(0);
        __syncthreads();   // buf[cur] published; buf[1-cur] free to refill

        if (t + 1 < NTILES) {
            if (wave == 0)
                tdm_load_g2_tile(g2 + (size_t)(m0 + MT) * E, cur ? ldsA : ldsB);
            if (tid < MT) {
                const int mm = m0 + MT + tid;
                const float a = x2[0 * NPTS + mm];
                const float b = x2[1 * NPTS + mm];
                const float c = x2[2 * NPTS + mm];
                (cur ? p2sA : p2sB)[tid] =
                    make_float4(a, b, c, a * a + b * b + c * c);
            }
        }

        // cosine GEMM: acc[16n x 16m] += g1_tile · g2_tileᵀ over E
        v8f acc = {0.f, 0.f, 0.f, 0.f, 0.f, 0.f, 0.f, 0.f};
        #pragma unroll
        for (int kk = 0; kk < E; kk += 4) {
            const int k = kk + khalf * 2;
            v2f a; a.x = g1s[row_a * ROWW + k]; a.y = g1s[row_a * ROWW + k + 1];
            v2f b; b.x = gb[mloc * ROWW + k];   b.y = gb[mloc * ROWW + k + 1];
            acc = __builtin_amdgcn_wmma_f32_16x16x4_f32(
                false, a, false, b, (short)0, acc, false, false);
        }

        // combine with exact squared-Euclid, write distance tile to LDS
        const float4 p2v = pb[mloc];
        #pragma unroll
        for (int r = 0; r < 8; ++r) {
            const int n = r + 8 * khalf;
            const float4 p1v = p1s[n];
            const float dotxy = p1v.x * p2v.x + p1v.y * p2v.y + p1v.z * p2v.z;
            ds[n * DSW + mloc] = p1v.w + p2v.w - 2.0f * dotxy + 1.0f - acc[r];
        }
        __syncthreads();

        // running top-16: 4 owners per row, 16 columns each
        if (tid < 64) {
            const int row = tid >> 2;
            const int seg = tid & 3;
            #pragma unroll
            for (int j = 0; j < 16; ++j) {
                const int ml = seg * 16 + j;
                const float v = ds[row * DSW + ml];
                if (v < bd[KNN - 1]) {
                    bd[KNN - 1] = v;
                    bi[KNN - 1] = m0 + ml;
                    #pragma unroll
                    for (int u = KNN - 1; u > 0; --u) {
                        if (bd[u] < bd[u - 1]) {
                            const float td = bd[u]; bd[u] = bd[u - 1]; bd[u - 1] = td;
                            const int   ti = bi[u]; bi[u] = bi[u - 1]; bi[u - 1] = ti;
                        }
                    }
                }
            }
        }
    }
    __syncthreads();

    // spill the 4 sorted partial lists per row
    if (tid < 64) {
        const int row = tid >> 2, seg = tid & 3;
        #pragma unroll
        for (int i = 0; i < KNN; ++i) {
            mrgd[(row * 4 + seg) * KNN + i] = bd[i];
            mrgi[(row * 4 + seg) * KNN + i] = bi[i];
        }
    }
    __syncthreads();

    // 4-way merge of sorted lists; ties -> smaller index (stable top_k match)
    if (tid < NT) {
        const int row = tid;
        int ptr[4] = {0, 0, 0, 0};
        for (int out = 0; out < KNN; ++out) {
            float bestv = 3.4e38f;
            int bestidx = 0x7fffffff;
            int bests = 0;
            #pragma unroll
            for (int s = 0; s < 4; ++s) {
                const int p = ptr[s];
                const float v = mrgd[(row * 4 + s) * KNN + p];
                const int  id = mrgi[(row * 4 + s) * KNN + p];
                if (v < bestv || (v == bestv && id < bestidx)) {
                    bestv = v; bestidx = id; bests = s;
                }
            }
            ptr[bests]++;
            idx_out[(size_t)(n0 + row) * KNN + out] = bestidx;
        }
    }
}

// ---------------------------------------------------------------------------
// Phase 3: gather neighbors, pos-conv, fused add+leaky, mlp0, leaky, maxpool.
// ---------------------------------------------------------------------------
__global__ __launch_bounds__(64) void out_kernel(
    const float* __restrict__ q1,   // [NPTS][C0]
    const float* __restrict__ q2,   // [NPTS][C0]
    const float* __restrict__ x1,   // [3][NPTS]
    const float* __restrict__ x2,   // [3][NPTS]
    const int* __restrict__ knn,    // [NPTS][KNN]
    const float* __restrict__ pos_w, const float* __restrict__ pos_b,
    const float* __restrict__ mlp_w, const float* __restrict__ mlp_b,
    float* __restrict__ out)        // [C1][NPTS]
{
    const int n = blockIdx.x;
    const int c = threadIdx.x;

    __shared__ float newf[KNN][C0 + 1];
    __shared__ float wls[C0][C0 + 1];
    __shared__ float q1s[C0];

    for (int i = c; i < C0 * C0; i += 64)
        wls[i / C0][i % C0] = mlp_w[i];
    q1s[c] = q1[(size_t)n * C0 + c];

    const float px = x1[0 * NPTS + n];
    const float py = x1[1 * NPTS + n];
    const float pz = x1[2 * NPTS + n];
    const float pwx = pos_w[c * 3 + 0];
    const float pwy = pos_w[c * 3 + 1];
    const float pwz = pos_w[c * 3 + 2];
    const float pb  = pos_b[c];
    __syncthreads();

    #pragma unroll
    for (int k = 0; k < KNN; ++k) {
        const int m = knn[(size_t)n * KNN + k];
        const float dx = x2[0 * NPTS + m] - px;
        const float dy = x2[1 * NPTS + m] - py;
        const float dz = x2[2 * NPTS + m] - pz;
        float v = q2[(size_t)m * C0 + c] + q1s[c]
                + (dx * pwx + dy * pwy + dz * pwz + pb);
        newf[k][c] = (v >= 0.f) ? v : 0.1f * v;
    }
    __syncthreads();

    const float bw = mlp_b[c];
    float best = -3.4e38f;
    for (int k = 0; k < KNN; ++k) {
        float acc = bw;
        #pragma unroll 8
        for (int cc = 0; cc < C0; ++cc)
            acc += wls[c][cc] * newf[k][cc];
        acc = (acc >= 0.f) ? acc : 0.1f * acc;
        best = fmaxf(best, acc);
    }
    out[(size_t)c * NPTS + n] = best;
}

// ---------------------------------------------------------------------------
extern "C" void kernel_launch(void* const* d_in, const int* in_sizes, int n_in,
                              void* d_out, int out_size, void* d_ws, size_t ws_size,
                              hipStream_t stream) {
    const float* pc1    = (const float*)d_in[0];
    const float* pc2    = (const float*)d_in[1];
    const float* feat1  = (const float*)d_in[2];
    const float* feat2  = (const float*)d_in[3];
    const float* t11_w  = (const float*)d_in[4];
    const float* t11_b  = (const float*)d_in[5];
    const float* t22_w  = (const float*)d_in[6];
    const float* t22_b  = (const float*)d_in[7];
    const float* pos_w  = (const float*)d_in[8];
    const float* pos_b  = (const float*)d_in[9];
    const float* dist_w = (const float*)d_in[10];
    const float* dist_b = (const float*)d_in[11];
    const float* mlp0_w = (const float*)d_in[12];
    const float* mlp0_b = (const float*)d_in[13];

    // workspace: q (4*N*C0 f32) | g (4*N*E f32) | knn (2*N*16 i32)  (~25 MB)
    float* q  = (float*)d_ws;
    float* g  = q + (size_t)4 * NPTS * C0;
    int*  knn = (int*)(g + (size_t)4 * NPTS * E);

    proj_kernel<<<dim3(NPTS / 4, 1, 4), 256, 0, stream>>>(
        feat1, feat2, t11_w, t11_b, t22_w, t22_b, q);

    gvec_kernel<<<dim3(NPTS, 4), 128, 0, stream>>>(q, dist_w, dist_b, g);

    // direction 0: rows from pc1/f1a, candidates from pc2/f2a
    dist_topk_kernel<<<NPTS / NT, 128, L_TOTAL, stream>>>(
        g, g + (size_t)NPTS * E, pc1, pc2, knn);
    // direction 1: rows from pc2/f2b, candidates from pc1/f1b
    dist_topk_kernel<<<NPTS / NT, 128, L_TOTAL, stream>>>(
        g + (size_t)2 * NPTS * E, g + (size_t)3 * NPTS * E, pc2, pc1,
        knn + (size_t)NPTS * KNN);

    float* out = (float*)d_out;
    out_kernel<<<NPTS, 64, 0, stream>>>(
        q, q + (size_t)NPTS * C0, pc1, pc2, knn,
        pos_w, pos_b, mlp0_w, mlp0_b, out);
    out_kernel<<<NPTS, 64, 0, stream>>>(
        q + (size_t)2 * NPTS * C0, q + (size_t)3 * NPTS * C0, pc2, pc1,
        knn + (size_t)NPTS * KNN,
        pos_w, pos_b, mlp0_w, mlp0_b, out + (size_t)C0 * NPTS);
}